// GCALayer_32839319945344
// MI455X (gfx1250) — compile-verified
//
#include <hip/hip_runtime.h>

typedef __attribute__((ext_vector_type(2))) float v2f;
typedef __attribute__((ext_vector_type(8))) float v8f;

#define NNODES 50000
#define NEDGES 800000
#define IN_F   128
#define TOPO_F 64
#define HC     64      // H*C
#define NEG_SLOPE 0.2f

// ---------- helpers: order-preserving float<->uint keys for atomic max ----------
__device__ __forceinline__ unsigned f2key(float f) {
    unsigned u = __float_as_uint(f);
    return (u & 0x80000000u) ? ~u : (u | 0x80000000u);
}
__device__ __forceinline__ float key2f(unsigned k) {
    unsigned u = (k & 0x80000000u) ? (k & 0x7fffffffu) : ~k;
    return __uint_as_float(u);
}

// ---------------------------------------------------------------------------
// Kernel 1: Y[M x 64] = X[M x K] @ W^T + b   (W is [64 x K], row-major)
//
// W is pre-permuted into LDS in exact per-lane WMMA B-fragment order
// (one ds_load_b64 per fragment in the inner loop — no global-load stall
// before each WMMA).  Each wave computes TWO 16-row tiles so every B
// fragment feeds 2 independent WMMA chains.
//
// f32 WMMA A layout: lanes 0-15 -> M rows, float2 = K {k0,k0+1};
//                    lanes 16-31 -> same rows, K {k0+2,k0+3}.
// B fragment mirrors it with lane -> N column.
// ---------------------------------------------------------------------------
__global__ __launch_bounds__(256) void gemm_wmma_f32(
    const float* __restrict__ X, const float* __restrict__ W,
    const float* __restrict__ bvec, float* __restrict__ Y, int M, int K)
{
    __shared__ v2f wfrag[(IN_F / 4) * 128];       // 4096 v2f = 32 KB max (K=128)

    const int lane = threadIdx.x & 31;
    const int wave = threadIdx.x >> 5;
    const int ntiles = M >> 4;                    // M multiple of 16
    const int t0 = blockIdx.x * 16 + wave;        // first 16-row tile
    const int t1 = t0 + 8;                        // second 16-row tile
    const bool v0 = (t0 < ntiles);
    const bool v1 = (t1 < ntiles);

    // ---- stage W into LDS as B fragments: wfrag[s*128 + nt*32 + lane] ----
    const int nsteps = K >> 2;
    const int nslots = nsteps * 128;
    for (int slot = threadIdx.x; slot < nslots; slot += 256) {
        int s   = slot >> 7;                      // k-step
        int rem = slot & 127;
        int nt  = rem >> 5;                       // n-tile
        int l   = rem & 31;
        int n   = nt * 16 + (l & 15);
        int k   = s * 4 + (l >> 4) * 2;
        v2f w;
        w[0] = W[(size_t)n * K + k];
        w[1] = W[(size_t)n * K + k + 1];
        wfrag[slot] = w;
    }
    __syncthreads();
    if (!v0 && !v1) return;

    const int hi  = lane >> 4;
    const int l15 = lane & 15;
    const int ar0 = (v0 ? t0 : 0) * 16 + l15;     // clamp invalid tiles to row 0
    const int ar1 = (v1 ? t1 : 0) * 16 + l15;
    const float* xrow0 = X + (size_t)ar0 * K + hi * 2;
    const float* xrow1 = X + (size_t)ar1 * K + hi * 2;

    v8f accA[4] = {};                             // tile t0: 4 x (16x16)
    v8f accB[4] = {};                             // tile t1

    for (int s = 0; s < nsteps; ++s) {
        v2f a0 = *(const v2f*)(xrow0 + s * 4);
        v2f a1 = *(const v2f*)(xrow1 + s * 4);
        const v2f* bf = wfrag + s * 128 + lane;
        v2f b0 = bf[0];
        v2f b1 = bf[32];
        v2f b2 = bf[64];
        v2f b3 = bf[96];
        accA[0] = __builtin_amdgcn_wmma_f32_16x16x4_f32(false, a0, false, b0, (short)0, accA[0], false, false);
        accB[0] = __builtin_amdgcn_wmma_f32_16x16x4_f32(false, a1, false, b0, (short)0, accB[0], false, false);
        accA[1] = __builtin_amdgcn_wmma_f32_16x16x4_f32(false, a0, false, b1, (short)0, accA[1], false, false);
        accB[1] = __builtin_amdgcn_wmma_f32_16x16x4_f32(false, a1, false, b1, (short)0, accB[1], false, false);
        accA[2] = __builtin_amdgcn_wmma_f32_16x16x4_f32(false, a0, false, b2, (short)0, accA[2], false, false);
        accB[2] = __builtin_amdgcn_wmma_f32_16x16x4_f32(false, a1, false, b2, (short)0, accB[2], false, false);
        accA[3] = __builtin_amdgcn_wmma_f32_16x16x4_f32(false, a0, false, b3, (short)0, accA[3], false, false);
        accB[3] = __builtin_amdgcn_wmma_f32_16x16x4_f32(false, a1, false, b3, (short)0, accB[3], false, false);
    }

    // C/D layout: VGPR v, lanes 0-15 -> row v, col=lane; lanes 16-31 -> row v+8.
    float bval[4];
#pragma unroll
    for (int nt = 0; nt < 4; ++nt) bval[nt] = bvec[nt * 16 + l15];

    if (v0) {
        const int rbase = t0 * 16 + hi * 8;
#pragma unroll
        for (int v = 0; v < 8; ++v) {
            float* yr = Y + (size_t)(rbase + v) * HC + l15;
#pragma unroll
            for (int nt = 0; nt < 4; ++nt) yr[nt * 16] = accA[nt][v] + bval[nt];
        }
    }
    if (v1) {
        const int rbase = t1 * 16 + hi * 8;
#pragma unroll
        for (int v = 0; v < 8; ++v) {
            float* yr = Y + (size_t)(rbase + v) * HC + l15;
#pragma unroll
            for (int nt = 0; nt < 4; ++nt) yr[nt * 16] = accB[nt][v] + bval[nt];
        }
    }
}

// ---------------------------------------------------------------------------
// Kernel 2: per-edge raw attention logit + global per-head max (atomic, keyed)
// ---------------------------------------------------------------------------
__global__ __launch_bounds__(256) void edge_alpha_max(
    const float* __restrict__ xh, const float* __restrict__ th,
    const int* __restrict__ ei, const float* __restrict__ att_node,
    const float* __restrict__ att_topo, float* __restrict__ alpha_raw,
    unsigned* __restrict__ maxkey)
{
    int e = blockIdx.x * blockDim.x + threadIdx.x;
    float a0 = -3.402823466e38f, a1 = -3.402823466e38f;
    if (e < NEDGES) {
        int cj = ei[NEDGES + e];                  // col
        const float* xv = xh + (size_t)cj * HC;
        const float* tv = th + (size_t)cj * HC;
        float s0 = 0.f, s1 = 0.f;
#pragma unroll 8
        for (int c = 0; c < 32; ++c) {
            s0 += xv[c]      * att_node[c]      + tv[c]      * att_topo[c];
            s1 += xv[32 + c] * att_node[32 + c] + tv[32 + c] * att_topo[32 + c];
        }
        a0 = (s0 > 0.f) ? s0 : NEG_SLOPE * s0;
        a1 = (s1 > 0.f) ? s1 : NEG_SLOPE * s1;
        alpha_raw[2 * e]     = a0;
        alpha_raw[2 * e + 1] = a1;
    }
#pragma unroll
    for (int off = 16; off; off >>= 1) {
        a0 = fmaxf(a0, __shfl_xor(a0, off, 32));
        a1 = fmaxf(a1, __shfl_xor(a1, off, 32));
    }
    if ((threadIdx.x & 31) == 0) {
        atomicMax(&maxkey[0], f2key(a0));
        atomicMax(&maxkey[1], f2key(a1));
    }
}

// ---------------------------------------------------------------------------
// Kernel 3: p = exp(alpha - max); atomic-sum Z per head (wave-reduced);
// coalesced scatter of unnormalized exp * x_j: each wave owns 32 edges,
// broadcasts (row,col,p0,p1) per edge via shuffles, lanes cover 64 channels.
// ---------------------------------------------------------------------------
__global__ __launch_bounds__(256) void edge_scatter(
    const float* __restrict__ xh, const float* __restrict__ alpha_raw,
    const int* __restrict__ ei, const unsigned* __restrict__ maxkey,
    float* __restrict__ Zsum, float* __restrict__ out)
{
    const int lane = threadIdx.x & 31;
    const int waveId = (blockIdx.x * blockDim.x + threadIdx.x) >> 5;
    const int e = waveId * 32 + lane;

    const float m0 = key2f(maxkey[0]);
    const float m1 = key2f(maxkey[1]);

    float p0 = 0.f, p1 = 0.f;
    int r = 0, cj = 0;
    if (e < NEDGES) {
        p0 = __expf(alpha_raw[2 * e]     - m0);
        p1 = __expf(alpha_raw[2 * e + 1] - m1);
        r  = ei[e];                               // row (dst)
        cj = ei[NEDGES + e];                      // col (src)
    }
    float s0 = p0, s1 = p1;
#pragma unroll
    for (int off = 16; off; off >>= 1) {
        s0 += __shfl_xor(s0, off, 32);
        s1 += __shfl_xor(s1, off, 32);
    }
    if (lane == 0) {
        atomicAdd(&Zsum[0], s0);
        atomicAdd(&Zsum[1], s1);
    }
#pragma unroll 4
    for (int i = 0; i < 32; ++i) {
        int   ri = __shfl(r, i, 32);
        int   ci = __shfl(cj, i, 32);
        float q0 = __shfl(p0, i, 32);
        float q1 = __shfl(p1, i, 32);
        if (q0 > 0.f) {                           // q0==0 only for tail-guard lanes
            const float* xv = xh + (size_t)ci * HC;
            float* ov = out + (size_t)ri * HC;
            atomicAdd(&ov[lane],      xv[lane]      * q0);   // head 0, ch=lane
            atomicAdd(&ov[32 + lane], xv[32 + lane] * q1);   // head 1
        }
    }
}

// ---------------------------------------------------------------------------
// Kernel 4: out = acc / Z_h + bias[ch]; copy topology passthrough output
// ---------------------------------------------------------------------------
__global__ __launch_bounds__(256) void finalize(
    float* __restrict__ out, const float* __restrict__ Zsum,
    const float* __restrict__ bias, const float* __restrict__ topo,
    float* __restrict__ out_topo)
{
    int i = blockIdx.x * blockDim.x + threadIdx.x;
    if (i < NNODES * HC) {
        int ch = i & 63;
        out[i] = out[i] / Zsum[ch >> 5] + bias[ch];
    }
    if (i < NNODES * TOPO_F) out_topo[i] = topo[i];   // same count: 3.2M
}

// ---------------------------------------------------------------------------
// Kernel 0: zero the accumulation region of d_out, init max keys / sums
// ---------------------------------------------------------------------------
__global__ __launch_bounds__(256) void init_ws(
    float* __restrict__ out, unsigned* __restrict__ maxkey, float* __restrict__ Zsum)
{
    int i = blockIdx.x * blockDim.x + threadIdx.x;
    if (i < NNODES * HC) out[i] = 0.f;
    if (i == 0) {
        maxkey[0] = f2key(-3.402823466e38f);
        maxkey[1] = f2key(-3.402823466e38f);
        Zsum[0] = 0.f;
        Zsum[1] = 0.f;
    }
}

extern "C" void kernel_launch(void* const* d_in, const int* in_sizes, int n_in,
                              void* d_out, int out_size, void* d_ws, size_t ws_size,
                              hipStream_t stream)
{
    (void)in_sizes; (void)n_in; (void)out_size; (void)ws_size;

    const float* x        = (const float*)d_in[0];
    const int*   ei       = (const int*)  d_in[1];   // (2,E) int32
    const float* topo     = (const float*)d_in[2];
    const float* W_lin    = (const float*)d_in[3];   // (64,128)
    const float* b_lin    = (const float*)d_in[4];
    const float* W_topo   = (const float*)d_in[5];   // (64,64)
    const float* b_topo   = (const float*)d_in[6];
    const float* att_node = (const float*)d_in[7];   // (1,2,32) -> 64 flat
    const float* att_topo = (const float*)d_in[8];
    const float* bias     = (const float*)d_in[9];

    float* out      = (float*)d_out;                 // (N,64)
    float* out_topo = out + (size_t)NNODES * HC;     // (N,64) passthrough

    // workspace layout
    float*    xh     = (float*)d_ws;                        // N*64
    float*    th     = xh + (size_t)NNODES * HC;            // N*64
    float*    alpha  = th + (size_t)NNODES * HC;            // E*2
    unsigned* maxkey = (unsigned*)(alpha + (size_t)NEDGES * 2);
    float*    Zsum   = (float*)(maxkey + 2);

    const int nOut = NNODES * HC;                    // 3,200,000

    init_ws<<<(nOut + 255) / 256, 256, 0, stream>>>(out, maxkey, Zsum);

    // xh = x @ W_lin^T + b_lin   (K=128); th = topo @ W_topo^T + b_topo (K=64)
    const int tiles = NNODES / 16;                   // 3125 (exact)
    const int gblk  = (tiles + 15) / 16;             // 8 waves x 2 tiles per block
    gemm_wmma_f32<<<gblk, 256, 0, stream>>>(x,    W_lin,  b_lin,  xh, NNODES, IN_F);
    gemm_wmma_f32<<<gblk, 256, 0, stream>>>(topo, W_topo, b_topo, th, NNODES, TOPO_F);

    edge_alpha_max<<<(NEDGES + 255) / 256, 256, 0, stream>>>(
        xh, th, ei, att_node, att_topo, alpha, maxkey);

    edge_scatter<<<(NEDGES + 255) / 256, 256, 0, stream>>>(
        xh, alpha, ei, maxkey, Zsum, out);

    finalize<<<(nOut + 255) / 256, 256, 0, stream>>>(out, Zsum, bias, topo, out_topo);
}